// Downsample_90297392431678
// MI455X (gfx1250) — compile-verified
//
#include <hip/hip_runtime.h>
#include <hip/hip_bf16.h>

// Sizes from the reference
#define BATCH   16
#define CFEAT   128
#define NPTS    4096
#define KSAMP   1024
#define EMBK    20

typedef __attribute__((ext_vector_type(2))) float v2f;
typedef __attribute__((ext_vector_type(8))) float v8f;

// ---------------------------------------------------------------------------
// Kernel 1: farthest point sampling. One block per batch, 1024 threads,
// each thread owns 4 points in registers; xyz resident in LDS.
// ---------------------------------------------------------------------------
__global__ __launch_bounds__(1024) void fps_kernel(const float* __restrict__ xyz,
                                                   int* __restrict__ idx_out) {
    int b = blockIdx.x;
    __shared__ float lx[NPTS], ly[NPTS], lz[NPTS];
    __shared__ float rbd[32];
    __shared__ int   rbi[32];
    __shared__ int   cur_s;
    const float* xb = xyz + (size_t)b * 3 * NPTS;
    int tid = threadIdx.x, lane = tid & 31, wave = tid >> 5;

    for (int i = tid; i < NPTS; i += 1024) {
        lx[i] = xb[i];
        ly[i] = xb[NPTS + i];
        lz[i] = xb[2 * NPTS + i];
    }
    __syncthreads();

    float px[4], py[4], pz[4], dist[4];
    int   pn[4];
#pragma unroll
    for (int j = 0; j < 4; j++) {
        int p = tid + 1024 * j;
        pn[j] = p; px[j] = lx[p]; py[j] = ly[p]; pz[j] = lz[p];
        dist[j] = 3.4e38f;
    }
    int cur = 0;
    for (int t = 0; t < KSAMP; t++) {
        if (tid == 0) idx_out[b * KSAMP + t] = cur;
        float cx = lx[cur], cy = ly[cur], cz = lz[cur];
        float bd = -1.0f; int bi = 0x7fffffff;
#pragma unroll
        for (int j = 0; j < 4; j++) {
            float dx = px[j] - cx, dy = py[j] - cy, dz = pz[j] - cz;
            float d = dx * dx + dy * dy + dz * dz;
            dist[j] = fminf(dist[j], d);
            if (dist[j] > bd) { bd = dist[j]; bi = pn[j]; }   // ties -> lower idx
        }
        // wave argmax, first-index tiebreak (matches jnp.argmax)
        for (int off = 16; off; off >>= 1) {
            float od = __shfl_xor(bd, off, 32);
            int   oi = __shfl_xor(bi, off, 32);
            if (od > bd || (od == bd && oi < bi)) { bd = od; bi = oi; }
        }
        if (lane == 0) { rbd[wave] = bd; rbi[wave] = bi; }
        __syncthreads();
        if (tid == 0) {
            float md = rbd[0]; int mi = rbi[0];
            for (int w = 1; w < 32; w++)
                if (rbd[w] > md || (rbd[w] == md && rbi[w] < mi)) { md = rbd[w]; mi = rbi[w]; }
            cur_s = mi;
        }
        __syncthreads();
        cur = cur_s;
    }
}

// ---------------------------------------------------------------------------
// Kernel 2: gather sampled features -> Q (B,K,128); write out1 (dsxyz) and
// out2 (idx as float).
// ---------------------------------------------------------------------------
__global__ __launch_bounds__(128) void gather_kernel(const float* __restrict__ x,
                                                     const float* __restrict__ xyz,
                                                     const int* __restrict__ idx,
                                                     float* __restrict__ Q,
                                                     float* __restrict__ out_xyz,
                                                     float* __restrict__ out_idxf) {
    int bk = blockIdx.x;               // b*1024 + k
    int b = bk >> 10, k = bk & 1023;
    int i = idx[bk];
    int c = threadIdx.x;
    Q[(size_t)bk * CFEAT + c] = x[((size_t)b * CFEAT + c) * NPTS + i];
    if (c < 3)
        out_xyz[((size_t)b * 3 + c) * KSAMP + k] = xyz[((size_t)b * 3 + c) * NPTS + i];
    if (c == 0)
        out_idxf[bk] = (float)i;
}

// ---------------------------------------------------------------------------
// Kernel 3: transpose x (B,C,N) -> S (B,N,C) row-major points.
// ---------------------------------------------------------------------------
__global__ __launch_bounds__(256) void transpose_kernel(const float* __restrict__ x,
                                                        float* __restrict__ S) {
    size_t gid = (size_t)blockIdx.x * 256 + threadIdx.x;   // (b*C + c)*N + n
    int n = (int)(gid & (NPTS - 1));
    size_t t = gid >> 12;
    int c = (int)(t & (CFEAT - 1));
    int b = (int)(t >> 7);
    S[((size_t)b * NPTS + n) * CFEAT + c] = x[gid];
}

// ---------------------------------------------------------------------------
// Kernel 4: per-point squared norms |s|^2.
// ---------------------------------------------------------------------------
__global__ __launch_bounds__(256) void norms_kernel(const float* __restrict__ S,
                                                    float* __restrict__ sn) {
    int gid = blockIdx.x * 256 + threadIdx.x;              // b*N + n
    const float* row = S + (size_t)gid * CFEAT;
    float s = 0.f;
#pragma unroll 8
    for (int c = 0; c < CFEAT; c++) { float v = row[c]; s += v * v; }
    sn[gid] = s;
}

// ---------------------------------------------------------------------------
// Kernel 5: KNN. One block per (batch, 16-query tile); 16 waves.
// A fragments (query tile) hoisted to registers once; inner loop is
// 1 global_load_b64 + 1 V_WMMA_F32_16X16X4_F32 per K-step.
// ---------------------------------------------------------------------------
__global__ __launch_bounds__(512) void knn_kernel(const float* __restrict__ Q,
                                                  const float* __restrict__ S,
                                                  const float* __restrict__ sn,
                                                  const int* __restrict__ idx,
                                                  int* __restrict__ knn) {
    int blk = blockIdx.x;
    int b = blk >> 6, qt = blk & 63;                        // 64 query tiles of 16
    __shared__ float Qt[16][CFEAT];
    __shared__ float qn_s[16];
    __shared__ float dchunk[16][256];
    int tid = threadIdx.x, lane = tid & 31, wave = tid >> 5;

    const float* Qb = Q + ((size_t)(b * KSAMP + qt * 16)) * CFEAT;
    for (int i = tid; i < 16 * CFEAT; i += 512) Qt[i >> 7][i & 127] = Qb[i];
    if (tid < 16) qn_s[tid] = sn[b * NPTS + idx[b * KSAMP + qt * 16 + tid]];
    __syncthreads();

    const float* Sb  = S + (size_t)b * NPTS * CFEAT;
    const float* snb = sn + b * NPTS;
    int m = lane & 15, khalf = (lane >> 4) * 2, mbase = (lane >> 4) * 8;

    // Hoist loop-invariant A fragments (16 queries x K=128) into registers.
    v2f ar[32];
#pragma unroll
    for (int kf = 0; kf < 32; kf++) {
        ar[kf][0] = Qt[m][kf * 4 + khalf];
        ar[kf][1] = Qt[m][kf * 4 + khalf + 1];
    }

    // per-lane ascending top-20 over this lane's strided candidates
    float td[EMBK]; int ti[EMBK];
#pragma unroll
    for (int j = 0; j < EMBK; j++) { td[j] = 3.4e38f; ti[j] = 0x7fffffff; }
    int q = wave;                                           // query owned for selection

    for (int nc0 = 0; nc0 < NPTS; nc0 += 256) {
        // ---- compute phase (uniform EXEC): wave w -> points nc0 + w*16 ----
        {
            int nc = nc0 + wave * 16;
            const float* srow = Sb + (size_t)(nc + m) * CFEAT;
            v8f acc = {};
#pragma unroll
            for (int kf = 0; kf < 32; kf++) {
                v2f bb;
                bb[0] = srow[kf * 4 + khalf];
                bb[1] = srow[kf * 4 + khalf + 1];
                acc = __builtin_amdgcn_wmma_f32_16x16x4_f32(false, ar[kf], false, bb,
                                                            (short)0, acc, false, false);
            }
            float snv = snb[nc + m];
#pragma unroll
            for (int r = 0; r < 8; r++)
                dchunk[mbase + r][wave * 16 + m] = qn_s[mbase + r] + snv - 2.0f * acc[r];
        }
        __syncthreads();
        // ---- selection phase: wave q scans row q, 8 candidates per lane ----
#pragma unroll
        for (int j = 0; j < 8; j++) {
            int col = lane + 32 * j;
            float d = dchunk[q][col];
            int gi = nc0 + col;
            if (d < td[EMBK - 1]) {
                bool placed = false;
#pragma unroll
                for (int s = EMBK - 1; s > 0; s--) {
                    if (!placed) {
                        if (d < td[s - 1]) { td[s] = td[s - 1]; ti[s] = ti[s - 1]; }
                        else { td[s] = d; ti[s] = gi; placed = true; }
                    }
                }
                if (!placed) { td[0] = d; ti[0] = gi; }
            }
        }
        __syncthreads();
    }

    // merge 32 per-lane lists: 20 rounds of wave argmin over list heads
    int out_base = (b * KSAMP + qt * 16 + q) * EMBK;
    for (int sel = 0; sel < EMBK; sel++) {
        float bd = td[0]; int bi = ti[0]; int bl = lane;
        for (int off = 16; off; off >>= 1) {
            float od = __shfl_xor(bd, off, 32);
            int   oi = __shfl_xor(bi, off, 32);
            int   ol = __shfl_xor(bl, off, 32);
            if (od < bd || (od == bd && oi < bi)) { bd = od; bi = oi; bl = ol; }
        }
        if (bl == lane) {                                   // winner pops its head
#pragma unroll
            for (int s = 0; s < EMBK - 1; s++) { td[s] = td[s + 1]; ti[s] = ti[s + 1]; }
            td[EMBK - 1] = 3.4e38f;
        }
        if (lane == 0) knn[out_base + sel] = bi;
    }
}

// ---------------------------------------------------------------------------
// Kernel 6: edge-conv MLP. One block per query (B*1024 blocks, 8 waves).
// E(32x256, rows>=20 zero) in LDS; GEMM1 -> H1(32x128), GEMM2 -> 32x256,
// leaky-relu(0.2), max over rows 0..19, write out0 (B,256,1024).
// Loops interchanged so shared A/B fragments feed multiple accumulators.
// ---------------------------------------------------------------------------
__global__ __launch_bounds__(256) void edge_mlp_kernel(const float* __restrict__ Q,
                                                       const float* __restrict__ S,
                                                       const int* __restrict__ knn,
                                                       const float* __restrict__ W1,
                                                       const float* __restrict__ b1,
                                                       const float* __restrict__ W2,
                                                       const float* __restrict__ b2,
                                                       float* __restrict__ out0) {
    int bk = blockIdx.x;                                    // b*1024 + k
    int b = bk >> 10, k = bk & 1023;
    __shared__ float E[32][256];
    __shared__ float H1[32][128];
    __shared__ int   nbr_s[EMBK];
    int tid = threadIdx.x, lane = tid & 31, wave = tid >> 5;

    if (tid < EMBK) nbr_s[tid] = knn[bk * EMBK + tid];
    __syncthreads();

    const float* Qrow = Q + (size_t)bk * CFEAT;
    const float* Sb   = S + (size_t)b * NPTS * CFEAT;
    for (int i = tid; i < 32 * 128; i += 256) {
        int j = i >> 7, c = i & 127;
        if (j < EMBK) {
            float qv = Qrow[c];
            float nv = Sb[(size_t)nbr_s[j] * CFEAT + c];
            E[j][c]       = qv;                             // [q, nbr - q]
            E[j][c + 128] = nv - qv;
        } else {
            E[j][c] = 0.f; E[j][c + 128] = 0.f;
        }
    }
    __syncthreads();

    int m = lane & 15, khalf = (lane >> 4) * 2, mbase = (lane >> 4) * 8;

    // GEMM1: H1 = lrelu(E @ W1 + b1); wave w owns columns n0 = w*16.
    // One W1 (B) fragment feeds both M-tiles.
    {
        int n0 = wave * 16;
        float bias = b1[n0 + m];
        v8f acc0 = {}, acc1 = {};
        for (int kk = 0; kk < 256; kk += 4) {
            v2f bb, a0, a1;
            bb[0] = W1[(size_t)(kk + khalf) * 128 + n0 + m];
            bb[1] = W1[(size_t)(kk + khalf + 1) * 128 + n0 + m];
            a0[0] = E[m][kk + khalf];
            a0[1] = E[m][kk + khalf + 1];
            a1[0] = E[16 + m][kk + khalf];
            a1[1] = E[16 + m][kk + khalf + 1];
            acc0 = __builtin_amdgcn_wmma_f32_16x16x4_f32(false, a0, false, bb,
                                                         (short)0, acc0, false, false);
            acc1 = __builtin_amdgcn_wmma_f32_16x16x4_f32(false, a1, false, bb,
                                                         (short)0, acc1, false, false);
        }
#pragma unroll
        for (int r = 0; r < 8; r++) {
            float h0 = acc0[r] + bias;
            float h1 = acc1[r] + bias;
            h0 = h0 > 0.f ? h0 : 0.2f * h0;
            h1 = h1 > 0.f ? h1 : 0.2f * h1;
            H1[mbase + r][n0 + m]      = h0;
            H1[16 + mbase + r][n0 + m] = h1;
        }
    }
    __syncthreads();

    // GEMM2: 32x256 = H1(32x128) @ W2(128x256). Wave w owns N-tiles at
    // n0a = w*16 and n0b = (w+8)*16; A fragments shared across N-tiles,
    // B fragments shared across M-tiles -> 4 accumulators, 4 loads / 4 wmma.
    const float NEG = -3.4e38f;
    int n0a = wave * 16;
    int n0b = (wave + 8) * 16;
    v8f acc00 = {}, acc01 = {}, acc10 = {}, acc11 = {};
    for (int kk = 0; kk < 128; kk += 4) {
        v2f a0, a1, bb0, bb1;
        a0[0]  = H1[m][kk + khalf];
        a0[1]  = H1[m][kk + khalf + 1];
        a1[0]  = H1[16 + m][kk + khalf];
        a1[1]  = H1[16 + m][kk + khalf + 1];
        bb0[0] = W2[(size_t)(kk + khalf) * 256 + n0a + m];
        bb0[1] = W2[(size_t)(kk + khalf + 1) * 256 + n0a + m];
        bb1[0] = W2[(size_t)(kk + khalf) * 256 + n0b + m];
        bb1[1] = W2[(size_t)(kk + khalf + 1) * 256 + n0b + m];
        acc00 = __builtin_amdgcn_wmma_f32_16x16x4_f32(false, a0, false, bb0,
                                                      (short)0, acc00, false, false);
        acc10 = __builtin_amdgcn_wmma_f32_16x16x4_f32(false, a1, false, bb0,
                                                      (short)0, acc10, false, false);
        acc01 = __builtin_amdgcn_wmma_f32_16x16x4_f32(false, a0, false, bb1,
                                                      (short)0, acc01, false, false);
        acc11 = __builtin_amdgcn_wmma_f32_16x16x4_f32(false, a1, false, bb1,
                                                      (short)0, acc11, false, false);
    }
    {
        float biasA = b2[n0a + m];
        float biasB = b2[n0b + m];
        float cmaxA = NEG, cmaxB = NEG;
#pragma unroll
        for (int r = 0; r < 8; r++) {
            // M-tile 0: rows mbase+r (0..15), all < 20 -> valid
            float hA = acc00[r] + biasA; hA = hA > 0.f ? hA : 0.2f * hA;
            float hB = acc01[r] + biasB; hB = hB > 0.f ? hB : 0.2f * hB;
            cmaxA = fmaxf(cmaxA, hA);
            cmaxB = fmaxf(cmaxB, hB);
            // M-tile 1: rows 16 + mbase + r, valid only if < 20
            int row = 16 + mbase + r;
            float gA = acc10[r] + biasA; gA = gA > 0.f ? gA : 0.2f * gA;
            float gB = acc11[r] + biasB; gB = gB > 0.f ? gB : 0.2f * gB;
            if (row < EMBK) { cmaxA = fmaxf(cmaxA, gA); cmaxB = fmaxf(cmaxB, gB); }
        }
        cmaxA = fmaxf(cmaxA, __shfl_xor(cmaxA, 16, 32));
        cmaxB = fmaxf(cmaxB, __shfl_xor(cmaxB, 16, 32));
        if (lane < 16) {
            out0[((size_t)b * 256 + n0a + m) * KSAMP + k] = cmaxA;
            out0[((size_t)b * 256 + n0b + m) * KSAMP + k] = cmaxB;
        }
    }
}

// ---------------------------------------------------------------------------
extern "C" void kernel_launch(void* const* d_in, const int* in_sizes, int n_in,
                              void* d_out, int out_size, void* d_ws, size_t ws_size,
                              hipStream_t stream) {
    const float* x   = (const float*)d_in[0];   // (B,128,4096)
    const float* xyz = (const float*)d_in[1];   // (B,3,4096)
    const float* W1  = (const float*)d_in[2];   // (256,128)
    const float* b1  = (const float*)d_in[3];   // (128)
    const float* W2  = (const float*)d_in[4];   // (128,256)
    const float* b2  = (const float*)d_in[5];   // (256)

    float* out0 = (float*)d_out;                          // (B,256,1024)
    float* out1 = out0 + (size_t)BATCH * 256 * KSAMP;     // (B,3,1024)
    float* out2 = out1 + (size_t)BATCH * 3 * KSAMP;       // (B,1024) idx as float

    char* ws = (char*)d_ws;
    int*   idx_ws = (int*)ws;    ws += (size_t)BATCH * KSAMP * sizeof(int);
    float* Qws    = (float*)ws;  ws += (size_t)BATCH * KSAMP * CFEAT * sizeof(float);
    float* Sws    = (float*)ws;  ws += (size_t)BATCH * NPTS * CFEAT * sizeof(float);
    float* snws   = (float*)ws;  ws += (size_t)BATCH * NPTS * sizeof(float);
    int*   knnws  = (int*)ws;

    fps_kernel<<<BATCH, 1024, 0, stream>>>(xyz, idx_ws);
    gather_kernel<<<BATCH * KSAMP, 128, 0, stream>>>(x, xyz, idx_ws, Qws, out1, out2);
    transpose_kernel<<<(BATCH * CFEAT * NPTS) / 256, 256, 0, stream>>>(x, Sws);
    norms_kernel<<<(BATCH * NPTS) / 256, 256, 0, stream>>>(Sws, snws);
    knn_kernel<<<BATCH * 64, 512, 0, stream>>>(Qws, Sws, snws, idx_ws, knnws);
    edge_mlp_kernel<<<BATCH * KSAMP, 256, 0, stream>>>(Qws, Sws, knnws, W1, b1, W2, b2, out0);
}